// WHTConv2D_31035433681532
// MI455X (gfx1250) — compile-verified
//
#include <hip/hip_runtime.h>
#include <hip/hip_bf16.h>

// WHT-domain conv on gfx1250:
//   fwd 2D Hadamard (WMMA f32 16x16x4) -> per-pixel channel GEMM with exp(v)
//   scaling + soft threshold (WMMA) -> inv 2D Hadamard with fused 1/4096,
//   crop and residual add.
// The 64x64 Hadamard matrix is materialized once per workgroup in LDS so the
// GEMM inner loops are pure ds_load + v_wmma (no per-iteration popcount/select
// VALU chains).

typedef __attribute__((ext_vector_type(2))) float v2f;
typedef __attribute__((ext_vector_type(8))) float v8f;

__device__ __forceinline__ v8f wmma_f32_4(v2f a, v2f b, v8f c) {
  // (neg_a, A, neg_b, B, c_mod, C, reuse_a, reuse_b)
  return __builtin_amdgcn_wmma_f32_16x16x4_f32(false, a, false, b, (short)0, c,
                                               false, false);
}

#define LSTR 65  // padded LDS row stride (floats) -> conflict-free column reads

// ---------------------------------------------------------------------------
// Shared body for the two WHT kernels:  OUT = (s*H) * P * (s*H)
// 16 waves; wave (ti,tj) owns one 16x16 tile of the 64x64 result.
// ---------------------------------------------------------------------------
__device__ __forceinline__ void wht2d_body(const float* __restrict__ Hs,
                                           const float* __restrict__ P,
                                           float* __restrict__ T1, v8f& acc2) {
  const int wave = threadIdx.x >> 5;  // 0..15
  const int lane = threadIdx.x & 31;
  const int ti = wave >> 2, tj = wave & 3;
  const int m = lane & 15;
  const int hi = lane >> 4;
  const int kk = hi * 2;

  // T1 = P * H   (contract over w).  A row-pair is consecutive -> ds_load_b64.
  v8f acc = {};
#pragma unroll
  for (int k0 = 0; k0 < 64; k0 += 4) {
    int k = k0 + kk;
    v2f a, b;
    a.x = P[(ti * 16 + m) * LSTR + k];
    a.y = P[(ti * 16 + m) * LSTR + k + 1];
    b.x = Hs[k * LSTR + tj * 16 + m];
    b.y = Hs[(k + 1) * LSTR + tj * 16 + m];
    acc = wmma_f32_4(a, b, acc);
  }
#pragma unroll
  for (int r = 0; r < 8; ++r)
    T1[(ti * 16 + r + 8 * hi) * LSTR + tj * 16 + m] = acc[r];
  __syncthreads();

  // OUT = H * T1  (contract over h).  A pair consecutive in Hs -> ds_load_b64.
  acc2 = v8f{};
#pragma unroll
  for (int k0 = 0; k0 < 64; k0 += 4) {
    int k = k0 + kk;
    v2f a, b;
    a.x = Hs[(ti * 16 + m) * LSTR + k];
    a.y = Hs[(ti * 16 + m) * LSTR + k + 1];
    b.x = T1[k * LSTR + tj * 16 + m];
    b.y = T1[(k + 1) * LSTR + tj * 16 + m];
    acc2 = wmma_f32_4(a, b, acc2);
  }
}

// Fill the LDS Hadamard table: Hs[i][j] = scale * (-1)^popc(i&j)
__device__ __forceinline__ void fill_hadamard(float* __restrict__ Hs,
                                              float scale) {
  for (int idx = threadIdx.x; idx < 4096; idx += 512) {
    int i = idx >> 6, j = idx & 63;
    Hs[i * LSTR + j] = (__popc(i & j) & 1) ? -scale : scale;
  }
}

// ---------------------------------------------------------------------------
// Kernel 1: forward WHT.  One workgroup (16 waves) per (b,c) plane.
//   F2 = H * pad(x) * H,  written to ws as [plane][64*64]
// ---------------------------------------------------------------------------
__global__ __launch_bounds__(512) void k_fwd_wht(const float* __restrict__ x,
                                                 float* __restrict__ f2) {
  __shared__ float Hs[64 * LSTR];
  __shared__ float P[64 * LSTR];
  __shared__ float T1[64 * LSTR];
  const int plane = blockIdx.x;  // b*128 + c
  const float* xp = x + (size_t)plane * (56 * 56);

  fill_hadamard(Hs, 1.0f);
  for (int idx = threadIdx.x; idx < 4096; idx += 512) {
    int h = idx >> 6, w = idx & 63;
    P[h * LSTR + w] = (h < 56 && w < 56) ? xp[h * 56 + w] : 0.0f;
  }
  __syncthreads();

  v8f acc2;
  wht2d_body(Hs, P, T1, acc2);

  const int wave = threadIdx.x >> 5;
  const int lane = threadIdx.x & 31;
  const int ti = wave >> 2, tj = wave & 3;
  const int m = lane & 15, hi = lane >> 4;
  float* o = f2 + (size_t)plane * 4096;
#pragma unroll
  for (int r = 0; r < 8; ++r)
    o[(ti * 16 + r + 8 * hi) * 64 + tj * 16 + m] = acc2[r];
}

// ---------------------------------------------------------------------------
// Kernel 2: channel mix in WHT domain + soft threshold.
//   f5[b,o,s] = sum_{k=(p,i)} conv_w[p,o,i] * (exp(log_v[p,s]) * f2[b,i,s])
//   f6 = copysign(relu(|f5| - relu(T[o])), f5)
// One workgroup (8 waves, 256 thr) per (b, 16-wide spatial strip).
// A = conv_w (L2-resident 128KB, pair loads -> global_load_b64); B in LDS.
// ---------------------------------------------------------------------------
__global__ __launch_bounds__(256) void k_chan_mix(const float* __restrict__ f2,
                                                  const float* __restrict__ conv_w,
                                                  const float* __restrict__ log_v,
                                                  const float* __restrict__ T,
                                                  float* __restrict__ f6) {
  __shared__ float Bs[256 * 16];  // [k=p*128+i][n] with exp(v) folded in
  __shared__ float thr[128];
  const int b = blockIdx.x >> 8;
  const int s0 = (blockIdx.x & 255) * 16;
  const float* f2b = f2 + (size_t)b * 128 * 4096;

  for (int idx = threadIdx.x; idx < 256 * 16; idx += 256) {
    int k = idx >> 4, n = idx & 15;
    float vv = __expf(log_v[(k >> 7) * 4096 + s0 + n]);
    Bs[idx] = f2b[(size_t)(k & 127) * 4096 + s0 + n] * vv;
  }
  if (threadIdx.x < 128) thr[threadIdx.x] = fmaxf(T[threadIdx.x], 0.0f);
  __syncthreads();

  const int wave = threadIdx.x >> 5;  // output row tile 0..7 (o = wave*16..)
  const int lane = threadIdx.x & 31;
  const int m = lane & 15;
  const int hi = lane >> 4;
  const int kk = hi * 2;

  v8f acc = {};
  const int orow = wave * 16 + m;
#pragma unroll 4
  for (int k0 = 0; k0 < 256; k0 += 4) {
    int k = k0 + kk;
    v2f a, bf;
    // W[o][k] = conv_w[k>>7][o][k&127]; pairs never straddle the p boundary,
    // so (k, k+1) are consecutive dwords -> one global_load_b64.
    const float* wrow = conv_w + ((size_t)(k >> 7) * 128 + orow) * 128;
    a.x = wrow[k & 127];
    a.y = wrow[(k & 127) + 1];
    bf.x = Bs[k * 16 + m];
    bf.y = Bs[(k + 1) * 16 + m];
    acc = wmma_f32_4(a, bf, acc);
  }

  float* out = f6 + (size_t)b * 128 * 4096;
#pragma unroll
  for (int r = 0; r < 8; ++r) {
    int o = wave * 16 + r + 8 * hi;
    float v = acc[r];
    float mag = fabsf(v) - thr[o];
    out[(size_t)o * 4096 + s0 + m] = (mag > 0.0f) ? copysignf(mag, v) : 0.0f;
  }
}

// ---------------------------------------------------------------------------
// Kernel 3: inverse WHT with the 1/64 normalization folded into the LDS
// Hadamard table (H/64 on both sides -> 1/4096 total), fused crop + residual.
// ---------------------------------------------------------------------------
__global__ __launch_bounds__(512) void k_inv_wht(const float* __restrict__ f6,
                                                 const float* __restrict__ x,
                                                 float* __restrict__ y) {
  __shared__ float Hs[64 * LSTR];
  __shared__ float P[64 * LSTR];
  __shared__ float T1[64 * LSTR];
  const int plane = blockIdx.x;  // b*128 + o
  const float* fp = f6 + (size_t)plane * 4096;

  fill_hadamard(Hs, 1.0f / 64.0f);
  for (int idx = threadIdx.x; idx < 4096; idx += 512) {
    int h = idx >> 6, w = idx & 63;
    P[h * LSTR + w] = fp[h * 64 + w];
  }
  __syncthreads();

  v8f acc2;
  wht2d_body(Hs, P, T1, acc2);

  const int wave = threadIdx.x >> 5;
  const int lane = threadIdx.x & 31;
  const int ti = wave >> 2, tj = wave & 3;
  const int m = lane & 15, hi = lane >> 4;

  const float* xp = x + (size_t)plane * (56 * 56);
  float* yp = y + (size_t)plane * (56 * 56);
#pragma unroll
  for (int r = 0; r < 8; ++r) {
    int h = ti * 16 + r + 8 * hi;
    int w = tj * 16 + m;
    if (h < 56 && w < 56) yp[h * 56 + w] = acc2[r] + xp[h * 56 + w];
  }
}

// ---------------------------------------------------------------------------
extern "C" void kernel_launch(void* const* d_in, const int* in_sizes, int n_in,
                              void* d_out, int out_size, void* d_ws,
                              size_t ws_size, hipStream_t stream) {
  (void)in_sizes; (void)n_in; (void)out_size; (void)ws_size;
  const float* x = (const float*)d_in[0];        // [32][128][56][56]
  const float* conv_w = (const float*)d_in[1];   // [2][128][128]
  const float* log_v = (const float*)d_in[2];    // [2][64][64]
  const float* T = (const float*)d_in[3];        // [128]
  float* y = (float*)d_out;                      // [32][128][56][56]

  float* f2 = (float*)d_ws;                           // 32*128*4096 f32 (64MB)
  float* f6 = f2 + (size_t)32 * 128 * 4096;           // 32*128*4096 f32 (64MB)

  k_fwd_wht<<<32 * 128, 512, 0, stream>>>(x, f2);
  k_chan_mix<<<32 * 256, 256, 0, stream>>>(f2, conv_w, log_v, T, f6);
  k_inv_wht<<<32 * 128, 512, 0, stream>>>(f6, x, y);
}